// SpatioTemporalGNN_84387517432398
// MI455X (gfx1250) — compile-verified
//
#include <hip/hip_runtime.h>
#include <math.h>

// ---------------------------------------------------------------------------
// SpatioTemporal GNN for MI455X (gfx1250, wave32).
// Dense projections use V_WMMA_F32_16X16X4_F32 (one 16x16 tile per wave).
// Edge softmax/aggregation is L2-resident gather + f32 atomics.
// ---------------------------------------------------------------------------

typedef __attribute__((ext_vector_type(2))) float v2f;
typedef __attribute__((ext_vector_type(8))) float v8f;

static constexpr int S_ = 6, F_ = 32, H_ = 4, C_ = 64, B_ = 16, D_ = 64, OUT_ = 64, AH_ = 4;

__device__ inline void atomicMaxF(float* addr, float v) {
  // monotone bit trick: positives via signed max, negatives via unsigned min
  if (v >= 0.f) atomicMax((int*)addr, __float_as_int(v));
  else          atomicMin((unsigned int*)addr, __float_as_uint(v));
}

__global__ void fill_f32(float* __restrict__ p, float v, long n) {
  long i = (long)blockIdx.x * blockDim.x + threadIdx.x;
  if (i < n) p[i] = v;
}

// ---------------------------------------------------------------------------
// C[M,Nc] = A[M,K] @ B[K,Nc]  (row-major, fp32), WMMA f32 16x16x4.
// One wave per 16x16 output tile. K must be a multiple of 4.
// A layout per ISA: lane(l<16) holds A[row, k..k+1], lane(l>=16) A[row, k+2..k+3].
// B layout: lanes 0-15 hold K rows (v + 0..1), lanes 16-31 hold K rows (v + 2..3).
// C/D: vgpr v -> row (v + 8*half), col = lane%16.
// ---------------------------------------------------------------------------
__global__ void gemm_f32_wmma(const float* __restrict__ A, const float* __restrict__ Bm,
                              float* __restrict__ Cm, int M, int K, int Nc) {
  int wave = blockIdx.x * (blockDim.x >> 5) + (threadIdx.x >> 5);
  int lane = threadIdx.x & 31;
  int ntiles = Nc >> 4;
  int mtiles = (M + 15) >> 4;
  if (wave >= mtiles * ntiles) return;            // wave-uniform exit
  int mt = wave / ntiles, nt = wave % ntiles;
  int half = lane >> 4, l16 = lane & 15;
  int arow = mt * 16 + l16; if (arow >= M) arow = M - 1;   // clamp, keep EXEC full
  int bcol = nt * 16 + l16;
  const float* Ar = A + (size_t)arow * K;
  v8f acc = {};
  for (int k = 0; k < K; k += 4) {
    int ka = k + half * 2;
    v2f a, b;
    a[0] = Ar[ka];
    a[1] = Ar[ka + 1];
    b[0] = Bm[(size_t)ka * Nc + bcol];
    b[1] = Bm[(size_t)(ka + 1) * Nc + bcol];
    acc = __builtin_amdgcn_wmma_f32_16x16x4_f32(false, a, false, b, (short)0, acc,
                                                false, false);
  }
#pragma unroll
  for (int v = 0; v < 8; v++) {
    int r = mt * 16 + v + 8 * half;
    if (r < M) Cm[(size_t)r * Nc + bcol] = acc[v];
  }
}

// per-node attention logits: s_src[n,h] = <H[n,h,:], att_src[h,:]>, same for dst
__global__ void node_scores(const float* __restrict__ Hm, const float* __restrict__ as_,
                            const float* __restrict__ ad_, float* __restrict__ ssrc,
                            float* __restrict__ sdst, int N, int heads, int C) {
  int i = blockIdx.x * blockDim.x + threadIdx.x;
  if (i >= N * heads) return;
  int n = i / heads, h = i % heads;
  const float* hp = Hm + ((size_t)n * heads + h) * C;
  const float* asp = as_ + h * C;
  const float* adp = ad_ + h * C;
  float s0 = 0.f, s1 = 0.f;
  for (int c = 0; c < C; c++) { float hv = hp[c]; s0 += hv * asp[c]; s1 += hv * adp[c]; }
  ssrc[i] = s0; sdst[i] = s1;
}

// pass A: segment max of leaky-relu edge scores (self-loops appended implicitly)
__global__ void edge_max(const int* __restrict__ esrc, const int* __restrict__ edst,
                         const float* __restrict__ ssrc, const float* __restrict__ sdst,
                         float* __restrict__ emax, int E, int N, int heads) {
  long i = (long)blockIdx.x * blockDim.x + threadIdx.x;
  long tot = (long)(E + N) * heads;
  if (i >= tot) return;
  int e = (int)(i / heads), h = (int)(i % heads);
  int s = (e < E) ? esrc[e] : (e - E);
  int d = (e < E) ? edst[e] : (e - E);
  float sc = ssrc[s * heads + h] + sdst[d * heads + h];
  sc = (sc > 0.f) ? sc : 0.2f * sc;
  atomicMaxF(&emax[d * heads + h], sc);
}

// pass B: accumulate exp(score - max) and exp * h[src] into dst (normalize later)
__global__ void edge_agg(const int* __restrict__ esrc, const int* __restrict__ edst,
                         const float* __restrict__ ssrc, const float* __restrict__ sdst,
                         const float* __restrict__ emax, float* __restrict__ esum,
                         const float* __restrict__ Hm, float* __restrict__ accum,
                         int E, int N, int heads, int C) {
  long i = (long)blockIdx.x * blockDim.x + threadIdx.x;
  long tot = (long)(E + N) * heads;
  if (i >= tot) return;
  int e = (int)(i / heads), h = (int)(i % heads);
  int s = (e < E) ? esrc[e] : (e - E);
  int d = (e < E) ? edst[e] : (e - E);
  const float4* hp = (const float4*)(Hm + ((size_t)s * heads + h) * C);
  __builtin_prefetch(hp, 0, 0);                   // global_prefetch: cover latency
  float sc = ssrc[s * heads + h] + sdst[d * heads + h];
  sc = (sc > 0.f) ? sc : 0.2f * sc;
  float ex = __expf(sc - emax[d * heads + h]);
  atomicAdd(&esum[d * heads + h], ex);
  float* ap = accum + ((size_t)d * heads + h) * C;
  for (int c4 = 0; c4 < (C >> 2); c4++) {
    float4 hv = hp[c4];
    atomicAdd(&ap[4 * c4 + 0], ex * hv.x);
    atomicAdd(&ap[4 * c4 + 1], ex * hv.y);
    atomicAdd(&ap[4 * c4 + 2], ex * hv.z);
    atomicAdd(&ap[4 * c4 + 3], ex * hv.w);
  }
}

// out[n,h,c] = accum/esum + bias   (+ optional relu)
__global__ void finalize_agg(const float* __restrict__ accum, const float* __restrict__ esum,
                             const float* __restrict__ bias, float* __restrict__ outp,
                             int N, int heads, int C, int do_relu) {
  long i = (long)blockIdx.x * blockDim.x + threadIdx.x;
  long tot = (long)N * heads * C;
  if (i >= tot) return;
  int hc = heads * C;
  int n = (int)(i / hc), j = (int)(i % hc);
  int h = j / C;
  float v = accum[i] / esum[n * heads + h] + bias[j];
  if (do_relu) v = fmaxf(v, 0.f);
  outp[i] = v;
}

__global__ void count_nodes(const int* __restrict__ batch, float* __restrict__ cnt, int N) {
  int i = blockIdx.x * blockDim.x + threadIdx.x;
  if (i < N) atomicAdd(&cnt[batch[i]], 1.0f);
}

__global__ void pool_sum(const float* __restrict__ out2, const int* __restrict__ batch,
                         float* __restrict__ pooled, int N, int t) {
  long i = (long)blockIdx.x * blockDim.x + threadIdx.x;
  if (i >= (long)N * D_) return;
  int n = (int)(i >> 6), c = (int)(i & 63);
  atomicAdd(&pooled[((size_t)batch[n] * S_ + t) * D_ + c], out2[i]);
}

__global__ void t1_x(const float* __restrict__ pooled, const float* __restrict__ cnt,
                     float* __restrict__ xattn) {
  int i = blockIdx.x * blockDim.x + threadIdx.x;
  if (i >= B_ * S_ * D_) return;
  int b = i / (S_ * D_);
  xattn[i] = pooled[i] / fmaxf(cnt[b], 1.0f);
}

__global__ void t1_qkv(const float* __restrict__ xattn, const float* __restrict__ Win,
                       const float* __restrict__ bin, float* __restrict__ qkv) {
  int i = blockIdx.x * blockDim.x + threadIdx.x;
  if (i >= B_ * S_ * 3 * D_) return;
  int bs = i / (3 * D_), j = i % (3 * D_);
  const float* xp = xattn + bs * D_;
  const float* wp = Win + j * D_;
  float acc = bin[j];
  for (int d = 0; d < D_; d++) acc += xp[d] * wp[d];
  qkv[i] = acc;
}

// one thread per (batch, head): 6x6 attention
__global__ void t2_attn(const float* __restrict__ qkv, float* __restrict__ o1) {
  int i = threadIdx.x;
  if (i >= B_ * AH_) return;
  int b = i / AH_, h = i % AH_;
  const int dh = D_ / AH_;                         // 16
  float sc[S_][S_];
  for (int qi = 0; qi < S_; qi++) {
    const float* qp = qkv + (b * S_ + qi) * 3 * D_ + h * dh;
    for (int kj = 0; kj < S_; kj++) {
      const float* kp = qkv + (b * S_ + kj) * 3 * D_ + D_ + h * dh;
      float s = 0.f;
      for (int d = 0; d < dh; d++) s += qp[d] * kp[d];
      sc[qi][kj] = s * 0.25f;                      // 1/sqrt(16)
    }
  }
  for (int qi = 0; qi < S_; qi++) {
    float m = sc[qi][0];
    for (int j = 1; j < S_; j++) m = fmaxf(m, sc[qi][j]);
    float sum = 0.f;
    for (int j = 0; j < S_; j++) { sc[qi][j] = __expf(sc[qi][j] - m); sum += sc[qi][j]; }
    float inv = 1.f / sum;
    for (int j = 0; j < S_; j++) sc[qi][j] *= inv;
  }
  for (int qi = 0; qi < S_; qi++)
    for (int d = 0; d < dh; d++) {
      float o = 0.f;
      for (int j = 0; j < S_; j++)
        o += sc[qi][j] * qkv[(b * S_ + j) * 3 * D_ + 2 * D_ + h * dh + d];
      o1[(b * S_ + qi) * D_ + h * dh + d] = o;
    }
}

// out-proj + residual + layernorm + temporal mean + final projection
__global__ void t3_final(const float* __restrict__ xattn, const float* __restrict__ o1,
                         const float* __restrict__ Wout, const float* __restrict__ bout,
                         const float* __restrict__ lng, const float* __restrict__ lnb,
                         const float* __restrict__ Wp, const float* __restrict__ bp,
                         float* __restrict__ outp) {
  __shared__ float y[B_ * S_ * D_];                // 24 KB
  __shared__ float ym[B_ * D_];                    // 4 KB
  int tid = threadIdx.x;
  for (int i = tid; i < B_ * S_ * D_; i += blockDim.x) {
    int bs = i / D_, d = i % D_;
    const float* op = o1 + bs * D_;
    const float* wp = Wout + d * D_;
    float acc = bout[d];
    for (int dd = 0; dd < D_; dd++) acc += op[dd] * wp[dd];
    y[i] = xattn[i] + acc;
  }
  __syncthreads();
  for (int bs = tid; bs < B_ * S_; bs += blockDim.x) {
    float* yp = y + bs * D_;
    float mu = 0.f;
    for (int d = 0; d < D_; d++) mu += yp[d];
    mu *= (1.f / D_);
    float var = 0.f;
    for (int d = 0; d < D_; d++) { float t = yp[d] - mu; var += t * t; }
    var *= (1.f / D_);
    float rinv = rsqrtf(var + 1e-5f);
    for (int d = 0; d < D_; d++) yp[d] = (yp[d] - mu) * rinv * lng[d] + lnb[d];
  }
  __syncthreads();
  for (int i = tid; i < B_ * D_; i += blockDim.x) {
    int b = i / D_, d = i % D_;
    float acc = 0.f;
    for (int s = 0; s < S_; s++) acc += y[(b * S_ + s) * D_ + d];
    ym[i] = acc * (1.f / S_);
  }
  __syncthreads();
  for (int i = tid; i < B_ * OUT_; i += blockDim.x) {
    int b = i / OUT_, o = i % OUT_;
    const float* wp = Wp + o * D_;
    float acc = bp[o];
    for (int d = 0; d < D_; d++) acc += ym[b * D_ + d] * wp[d];
    outp[i] = acc;
  }
}

extern "C" void kernel_launch(void* const* d_in, const int* in_sizes, int n_in,
                              void* d_out, int out_size, void* d_ws, size_t ws_size,
                              hipStream_t stream) {
  const float* x_seq = (const float*)d_in[0];
  const int*   eidx  = (const int*)d_in[1];
  const int*   batch = (const int*)d_in[2];
  const float* W1    = (const float*)d_in[3];
  const float* as1   = (const float*)d_in[4];
  const float* ad1   = (const float*)d_in[5];
  const float* b1    = (const float*)d_in[6];
  const float* W2    = (const float*)d_in[7];
  const float* as2   = (const float*)d_in[8];
  const float* ad2   = (const float*)d_in[9];
  const float* b2    = (const float*)d_in[10];
  const float* Win   = (const float*)d_in[11];
  const float* bin   = (const float*)d_in[12];
  const float* Wout  = (const float*)d_in[13];
  const float* bout  = (const float*)d_in[14];
  const float* lng   = (const float*)d_in[15];
  const float* lnb   = (const float*)d_in[16];
  const float* Wp    = (const float*)d_in[17];
  const float* bp    = (const float*)d_in[18];
  float* outp = (float*)d_out;

  const int N = in_sizes[2];
  const int E = in_sizes[1] / 2;
  const int* esrc = eidx;
  const int* edst = eidx + E;

  float* ws = (float*)d_ws;
  size_t off = 0;
  auto carve = [&](size_t nel) { float* p = ws + off; off += nel; return p; };
  float* H1    = carve((size_t)N * H_ * C_);
  float* out1  = carve((size_t)N * H_ * C_);
  float* H2v   = carve((size_t)N * C_);
  float* out2  = carve((size_t)N * C_);
  float* ssrc1 = carve((size_t)N * H_);
  float* sdst1 = carve((size_t)N * H_);
  float* ssrc2 = carve((size_t)N);
  float* sdst2 = carve((size_t)N);
  float* zgrp  = ws + off;                       // zeroed each timestep (contiguous)
  float* accum1 = carve((size_t)N * H_ * C_);
  float* esum1  = carve((size_t)N * H_);
  float* accum2 = carve((size_t)N * C_);
  float* esum2  = carve((size_t)N);
  const long zcnt = (long)N * (H_ * C_ + H_ + C_ + 1);
  float* igrp  = ws + off;                       // -inf each timestep
  float* emax1 = carve((size_t)N * H_);
  float* emax2 = carve((size_t)N);
  const long icnt = (long)N * (H_ + 1);
  float* ogrp  = ws + off;                       // zeroed once per launch
  float* pooled = carve((size_t)B_ * S_ * D_);
  float* cnt    = carve((size_t)B_);
  const long ocnt = (long)B_ * S_ * D_ + B_;
  float* xattn = carve((size_t)B_ * S_ * D_);
  float* qkv   = carve((size_t)B_ * S_ * 3 * D_);
  float* o1    = carve((size_t)B_ * S_ * D_);
  (void)ws_size; (void)n_in; (void)out_size;

  auto cdiv = [](long a, long b) { return (int)((a + b - 1) / b); };
  const float NEG_INF = -__builtin_huge_valf();

  fill_f32<<<cdiv(ocnt, 256), 256, 0, stream>>>(ogrp, 0.f, ocnt);
  count_nodes<<<cdiv(N, 256), 256, 0, stream>>>(batch, cnt, N);

  for (int t = 0; t < S_; t++) {
    const float* xt = x_seq + (size_t)t * N * F_;
    fill_f32<<<cdiv(zcnt, 256), 256, 0, stream>>>(zgrp, 0.f, zcnt);
    fill_f32<<<cdiv(icnt, 256), 256, 0, stream>>>(igrp, NEG_INF, icnt);

    // ---- GAT layer 1 ----
    {
      long tiles = (long)cdiv(N, 16) * (H_ * C_ / 16);
      gemm_f32_wmma<<<cdiv(tiles * 32, 128), 128, 0, stream>>>(xt, W1, H1, N, F_, H_ * C_);
    }
    node_scores<<<cdiv((long)N * H_, 256), 256, 0, stream>>>(H1, as1, ad1, ssrc1, sdst1, N, H_, C_);
    {
      long tot = (long)(E + N) * H_;
      edge_max<<<cdiv(tot, 256), 256, 0, stream>>>(esrc, edst, ssrc1, sdst1, emax1, E, N, H_);
      edge_agg<<<cdiv(tot, 256), 256, 0, stream>>>(esrc, edst, ssrc1, sdst1, emax1, esum1,
                                                   H1, accum1, E, N, H_, C_);
    }
    finalize_agg<<<cdiv((long)N * H_ * C_, 256), 256, 0, stream>>>(accum1, esum1, b1, out1,
                                                                   N, H_, C_, 1);
    // ---- GAT layer 2 ----
    {
      long tiles = (long)cdiv(N, 16) * (C_ / 16);
      gemm_f32_wmma<<<cdiv(tiles * 32, 128), 128, 0, stream>>>(out1, W2, H2v, N, H_ * C_, C_);
    }
    node_scores<<<cdiv((long)N, 256), 256, 0, stream>>>(H2v, as2, ad2, ssrc2, sdst2, N, 1, C_);
    {
      long tot = (long)(E + N);
      edge_max<<<cdiv(tot, 256), 256, 0, stream>>>(esrc, edst, ssrc2, sdst2, emax2, E, N, 1);
      edge_agg<<<cdiv(tot, 256), 256, 0, stream>>>(esrc, edst, ssrc2, sdst2, emax2, esum2,
                                                   H2v, accum2, E, N, 1, C_);
    }
    finalize_agg<<<cdiv((long)N * C_, 256), 256, 0, stream>>>(accum2, esum2, b2, out2, N, 1, C_, 0);
    pool_sum<<<cdiv((long)N * C_, 256), 256, 0, stream>>>(out2, batch, pooled, N, t);
  }

  // ---- temporal transformer head ----
  t1_x<<<cdiv(B_ * S_ * D_, 256), 256, 0, stream>>>(pooled, cnt, xattn);
  t1_qkv<<<cdiv(B_ * S_ * 3 * D_, 256), 256, 0, stream>>>(xattn, Win, bin, qkv);
  t2_attn<<<1, 64, 0, stream>>>(qkv, o1);
  t3_final<<<1, 256, 0, stream>>>(xattn, o1, Wout, bout, lng, lnb, Wp, bp, outp);
}